// CSENet_88665304858663
// MI455X (gfx1250) — compile-verified
//
#include <hip/hip_runtime.h>
#include <hip/hip_bf16.h>
#include <math.h>

typedef float v2f __attribute__((ext_vector_type(2)));
typedef float v8f __attribute__((ext_vector_type(8)));

#define DIMS   16
#define HW     12544      // 112*112
#define NPIX   50176      // 4 images * HW
#define NVERT  27554
#define VPAD   27568      // 1723 * 16
#define NCHUNK 1723
#define FEATN  (NPIX * DIMS)   // 802816

// ---------------------------------------------------------------------------
// c2[v] = sum_k E[v][k]^2 for v < NVERT, +INF padding up to VPAD so padded
// vertex columns can never win the argmin (INF - finite = INF).
// ---------------------------------------------------------------------------
__global__ void cse_c2_kernel(const float* __restrict__ emb,
                              float* __restrict__ c2) {
    int v = blockIdx.x * blockDim.x + threadIdx.x;
    if (v >= VPAD) return;
    float s = INFINITY;
    if (v < NVERT) {
        const float* r = emb + v * DIMS;
        s = 0.0f;
#pragma unroll
        for (int k = 0; k < DIMS; ++k) s += r[k] * r[k];
    }
    c2[v] = s;
}

// ---------------------------------------------------------------------------
// Fused distance-GEMM + argmin via V_WMMA_F32_16X16X4_F32 (exact f32).
//   dist = c2[v] - 2 * x . e_v   (x^2 constant per pixel, drops from argmin)
//
// Each wave owns TWO 16-pixel M-tiles (32 pixels): A for both tiles lives in
// 16 VGPRs for the whole kernel, and each 16-vertex chunk's 5 vector loads
// (4x B64 of E + c2) feed 8 WMMAs (two independent 4-deep K-chains), halving
// the E-stream bandwidth per pixel vs one tile/wave.
//
// The c2 broadcast is built once per chunk and used as the C operand of the
// FIRST WMMA of both chains (D != C is legal for WMMA), so the second tile's
// accumulator init costs zero extra movs.
//
// f32 16x16x4 A layout (ISA 7.12.2): lane L holds row M=L%16; per K-step kk,
// the two A VGPRs hold K = 4*kk + 2*(L/16) + {0,1}. B (4x16) mirrors with
// N = L%16. C/D: lane L, VGPR i -> M = i + 8*(L/16), N = L%16, so the
// candidate vertex per chunk is vBase + L%16 for all 8 rows a lane holds.
// ---------------------------------------------------------------------------
__global__ __launch_bounds__(128) void cse_argmin_kernel(
    const float* __restrict__ feats,
    const float* __restrict__ emb,
    const float* __restrict__ c2,
    float* __restrict__ out) {
    // Output 0: feats passthrough (reference returns (feats, dps)).
    for (int i = blockIdx.x * blockDim.x + threadIdx.x; i < FEATN;
         i += gridDim.x * blockDim.x)
        out[i] = feats[i];

    const int lane = threadIdx.x & 31;
    const int wave = threadIdx.x >> 5;
    const int half = lane >> 4;   // half-wave: selects K pair / M half
    const int mn   = lane & 15;   // M row for A, N col for B/C/D

    const int group = blockIdx.x * 4 + wave;  // 32-pixel group per wave
    const int gBase = group * 32;

    // ---- Load both A tiles (scaled by -2) into 2x4 float2 (16 VGPRs).
    // feats layout [4][16][112][112]; 12544 % 32 == 0 so a 32-pixel group
    // never straddles an image boundary.
    const int img   = gBase / HW;
    const int pBase = gBase - img * HW;
    const float* fb = feats + img * (DIMS * HW) + pBase;
    v2f a0[4], a1[4];
#pragma unroll
    for (int kk = 0; kk < 4; ++kk) {
        const int k0 = 4 * kk + 2 * half;
        a0[kk].x = -2.0f * fb[(k0 + 0) * HW + mn];
        a0[kk].y = -2.0f * fb[(k0 + 1) * HW + mn];
        a1[kk].x = -2.0f * fb[(k0 + 0) * HW + 16 + mn];
        a1[kk].y = -2.0f * fb[(k0 + 1) * HW + 16 + mn];
    }

    float best0[8], best1[8];
    int   bidx0[8], bidx1[8];
#pragma unroll
    for (int i = 0; i < 8; ++i) {
        best0[i] = INFINITY; bidx0[i] = 0;
        best1[i] = INFINITY; bidx1[i] = 0;
    }

    for (int chunk = 0; chunk < NCHUNK; ++chunk) {
        const int n  = chunk * 16 + mn;                // candidate vertex id
        const int nc = (n < NVERT) ? n : (NVERT - 1);  // clamp tail loads
        const float c2v = c2[n];                       // +INF in padded tail

        // B chunk: column n of E^T == row n of E (16 contiguous floats).
        const float* erow = emb + nc * DIMS;
        v2f bv[4];
#pragma unroll
        for (int kk = 0; kk < 4; ++kk)
            bv[kk] = *(const v2f*)(erow + 4 * kk + 2 * half);

        // One c2 broadcast shared by both chains' first WMMA as C operand.
        const v8f splat = {c2v, c2v, c2v, c2v, c2v, c2v, c2v, c2v};

        v8f acc0, acc1;
        acc0 = __builtin_amdgcn_wmma_f32_16x16x4_f32(false, a0[0], false, bv[0],
                                                     (short)0, splat, false, false);
        acc1 = __builtin_amdgcn_wmma_f32_16x16x4_f32(false, a1[0], false, bv[0],
                                                     (short)0, splat, false, false);
        acc0 = __builtin_amdgcn_wmma_f32_16x16x4_f32(false, a0[1], false, bv[1],
                                                     (short)0, acc0, false, false);
        acc1 = __builtin_amdgcn_wmma_f32_16x16x4_f32(false, a1[1], false, bv[1],
                                                     (short)0, acc1, false, false);
        acc0 = __builtin_amdgcn_wmma_f32_16x16x4_f32(false, a0[2], false, bv[2],
                                                     (short)0, acc0, false, false);
        acc1 = __builtin_amdgcn_wmma_f32_16x16x4_f32(false, a1[2], false, bv[2],
                                                     (short)0, acc1, false, false);
        acc0 = __builtin_amdgcn_wmma_f32_16x16x4_f32(false, a0[3], false, bv[3],
                                                     (short)0, acc0, false, false);
        acc1 = __builtin_amdgcn_wmma_f32_16x16x4_f32(false, a1[3], false, bv[3],
                                                     (short)0, acc1, false, false);

        // Running argmin update for both tiles (candidate index n is the
        // same for all 8 rows a lane holds).
#pragma unroll
        for (int i = 0; i < 8; ++i) {
            const bool lt0 = acc0[i] < best0[i];
            best0[i] = lt0 ? acc0[i] : best0[i];
            bidx0[i] = lt0 ? n : bidx0[i];
            const bool lt1 = acc1[i] < best1[i];
            best1[i] = lt1 ? acc1[i] : best1[i];
            bidx1[i] = lt1 ? n : bidx1[i];
        }
    }

    // ---- Cross-lane argmin per pixel row via LDS (per-wave regions).
    __shared__ float sval[8][16][16];
    __shared__ int   sidx[8][16][16];
#pragma unroll
    for (int i = 0; i < 8; ++i) {
        const int r = i + 8 * half;   // C/D row M for acc slot i
        sval[wave * 2 + 0][r][mn] = best0[i];
        sidx[wave * 2 + 0][r][mn] = bidx0[i];
        sval[wave * 2 + 1][r][mn] = best1[i];
        sidx[wave * 2 + 1][r][mn] = bidx1[i];
    }
    __syncthreads();

    // All 32 lanes reduce: half selects which tile, mn selects the row.
    {
        const int t = wave * 2 + half;
        float bv = sval[t][mn][0];
        int   bi = sidx[t][mn][0];
#pragma unroll
        for (int c = 1; c < 16; ++c) {
            const float v = sval[t][mn][c];
            const int   x = sidx[t][mn][c];
            if (v < bv) { bv = v; bi = x; }
        }
        // Output 1: argmin index (as float, per harness output dtype).
        out[FEATN + gBase + 16 * half + mn] = (float)bi;
    }
}

// ---------------------------------------------------------------------------
extern "C" void kernel_launch(void* const* d_in, const int* in_sizes, int n_in,
                              void* d_out, int out_size, void* d_ws,
                              size_t ws_size, hipStream_t stream) {
    (void)in_sizes; (void)n_in; (void)out_size; (void)ws_size;
    const float* feats = (const float*)d_in[0];
    const float* emb   = (const float*)d_in[1];
    float* out = (float*)d_out;
    float* c2  = (float*)d_ws;   // VPAD floats = 110 KB of scratch

    cse_c2_kernel<<<(VPAD + 255) / 256, 256, 0, stream>>>(emb, c2);
    // 392 blocks * 4 waves * 32 pixels = 50176 pixels, exact cover.
    cse_argmin_kernel<<<NPIX / 128, 128, 0, stream>>>(feats, emb, c2, out);
}